// MyBatchNormWeightMM_cuda_3813930959609
// MI455X (gfx1250) — compile-verified
//
#include <hip/hip_runtime.h>

#define LMAX    11
#define T_FRAG  96          // T_FF == T_O == 96
#define BATCH   2048
#define FLATFF  13824       // 96 * 144
#define NCHAN   1152        // 96 * 12
#define BN_EPS  1e-5f

typedef __attribute__((ext_vector_type(2))) float v2f;
typedef __attribute__((ext_vector_type(8))) float v8f;

// ---------------------------------------------------------------------------
// Phase 1: per-channel inverse BN scale.
// inv[l*96+t] = 1 / ( (bn[l*96+t] + sqrt(mean_{b,m}(Fr^2+Fi^2)) ) / 2 + eps )
// One 256-thread block per channel (1152 blocks). Reads FF once (226 MB).
// ---------------------------------------------------------------------------
__global__ __launch_bounds__(256)
void bn_stats_kernel(const float* __restrict__ FF,
                     const float* __restrict__ bn_stds,
                     float* __restrict__ inv_scale) {
    const int ch = blockIdx.x;            // 0..1151
    const int l  = ch / T_FRAG;
    const int t  = ch - l * T_FRAG;
    const int m  = 2 * l + 1;
    const long offF = (long)T_FRAG * l * l;          // row offset of degree l
    const float* base = FF + (offF + (long)t * m) * 2;
    const int total = BATCH * m;

    float acc = 0.f;
    for (int idx = threadIdx.x; idx < total; idx += 256) {
        int b = idx / m;
        int j = idx - b * m;
        const float2 v = *(const float2*)(base + ((long)b * FLATFF + j) * 2);
        acc += v.x * v.x + v.y * v.y;
    }

    __shared__ float red[256];
    red[threadIdx.x] = acc;
    __syncthreads();
    for (int s = 128; s > 0; s >>= 1) {
        if (threadIdx.x < s) red[threadIdx.x] += red[threadIdx.x + s];
        __syncthreads();
    }
    if (threadIdx.x == 0) {
        float mean    = red[0] / (float)total;
        float stdv    = sqrtf(mean);
        float new_std = (bn_stds[ch] + stdv) * 0.5f;   // BN_CNT = 1.0
        inv_scale[ch] = 1.0f / (new_std + BN_EPS);
    }
}

// ---------------------------------------------------------------------------
// Phase 2: segmented complex GEMM via V_WMMA_F32_16X16X4_F32.
//   out[(o), (b,j)] = sum_k A[o,k] * Bm[k,(b,j)],  K = 192 (re/im interleaved)
//   Ar[o,2c+0]=Wr[c,o]*inv  Ar[o,2c+1]=-Wi[c,o]*inv   -> real part
//   Ai[o,2c+0]=Wi[c,o]*inv  Ai[o,2c+1]= Wr[c,o]*inv   -> imag part
// LDS holds pre-built A fragments as float4 {P,-Q,Q,P} in [c][o] order:
//  - one ds_load_b128 per M-tile per K-step feeds two WMMAs, no VALU fixup
//  - 6 tile loads are 256 B apart -> all DS immediate offsets
//  - K-step advance is a uniform +3072 B, folded into immediates by unrolling
// Grid: 16*m blocks per degree (2304 total); block = 8 waves; wave = 96x16 tile.
// ---------------------------------------------------------------------------
__global__ __launch_bounds__(256)
void cwmm_wmma_kernel(const float* __restrict__ FF,
                      const float* __restrict__ W,
                      const float* __restrict__ inv_scale,
                      float* __restrict__ out) {
    // ---- block -> (degree l, column block) -------------------------------
    int bid = blockIdx.x;                       // 0 .. 16*144-1
    int l = 0;
    while (bid >= 16 * (l + 1) * (l + 1)) ++l;  // blocks before degree l: 16*l^2
    const int  nblk = bid - 16 * l * l;
    const int  m    = 2 * l + 1;
    const long offF = (long)T_FRAG * l * l;     // FF/out row offset for degree l
    const long offW = (long)(T_FRAG * T_FRAG) * l;

    // ---- stage A fragments {P,-Q,Q,P} into LDS, [c][o] linear ------------
    __shared__ float4 afrag[T_FRAG * T_FRAG];   // 144 KB
    for (int i = threadIdx.x; i < T_FRAG * T_FRAG; i += 256) {
        int c = i / T_FRAG;                     // i = c*96 + o -> coalesced
        float s = inv_scale[l * T_FRAG + c];
        const float2 w = *(const float2*)(W + (offW + i) * 2);
        const float P = w.x * s, Q = w.y * s;
        afrag[i] = make_float4(P, -Q, Q, P);
    }
    __syncthreads();

    // ---- per-wave geometry ----------------------------------------------
    const int wave  = threadIdx.x >> 5;
    const int lane  = threadIdx.x & 31;
    const int laneN = lane & 15;                // A row / B col / D col
    const int laneH = lane >> 4;                // selects K pair within 4-wide K

    const long g    = (long)nblk * 128 + wave * 16 + laneN;  // column = b*m + j
    const int  bcol = (int)(g / m);
    const int  jcol = (int)(g - (long)bcol * m);

    // B fragment source: F[(bcol), channel c = 2k+laneH, jcol], complex pair.
    const float2* bptr = (const float2*)FF
                       + ((long)bcol * FLATFF + offF + (long)laneH * m + jcol);
    const long bstep = 2L * m;                  // float2 stride per K-step (c += 2)

    // A fragment base in LDS: entry (c, o) at c*96 + o; c = 2k + laneH.
    const float4* abase = afrag + (long)laneH * T_FRAG + laneN;

    v8f accR[6], accI[6];
    #pragma unroll
    for (int t = 0; t < 6; ++t) {
        accR[t] = (v8f){0.f,0.f,0.f,0.f,0.f,0.f,0.f,0.f};
        accI[t] = (v8f){0.f,0.f,0.f,0.f,0.f,0.f,0.f,0.f};
    }

    // ---- K loop: 48 steps; per step 1 global b64 + 6 ds b128 + 12 WMMA ---
    #pragma unroll 4
    for (int k = 0; k < 48; ++k) {
        const float2 bv = bptr[(long)k * bstep];
        const v2f bfrag = {bv.x, bv.y};
        const float4* arow = abase + (long)k * (2 * T_FRAG);  // c = 2k + laneH
        #pragma unroll
        for (int t = 0; t < 6; ++t) {
            const float4 a = arow[t * 16];      // o = t*16 + laneN
            const v2f ar = { a.x, a.y };        // {P, -Q}
            const v2f ai = { a.z, a.w };        // {Q,  P}
            accR[t] = __builtin_amdgcn_wmma_f32_16x16x4_f32(
                          false, ar, false, bfrag, (short)0, accR[t], false, false);
            accI[t] = __builtin_amdgcn_wmma_f32_16x16x4_f32(
                          false, ai, false, bfrag, (short)0, accI[t], false, false);
        }
    }

    // ---- store: D lane holds col=laneN, row o = t*16 + v + 8*laneH -------
    // Output is write-once: non-temporal stores preserve L2 for FF/W.
    const long obase = (long)bcol * FLATFF + offF + jcol;   // + o*m per row
    v2f* op = (v2f*)out;
    #pragma unroll
    for (int t = 0; t < 6; ++t) {
        #pragma unroll
        for (int v = 0; v < 8; ++v) {
            const int o = t * 16 + v + 8 * laneH;
            const v2f val = { accR[t][v], accI[t][v] };
            __builtin_nontemporal_store(val, &op[obase + (long)o * m]);
        }
    }
}

// ---------------------------------------------------------------------------
extern "C" void kernel_launch(void* const* d_in, const int* in_sizes, int n_in,
                              void* d_out, int out_size, void* d_ws, size_t ws_size,
                              hipStream_t stream) {
    const float* FF = (const float*)d_in[0];   // [2048,13824,2]
    const float* W  = (const float*)d_in[1];   // [110592,2]
    const float* bn = (const float*)d_in[2];   // [1152]
    float* out = (float*)d_out;                // [2048,13824,2]
    float* inv = (float*)d_ws;                 // 1152 floats scratch

    bn_stats_kernel<<<NCHAN, 256, 0, stream>>>(FF, bn, inv);
    cwmm_wmma_kernel<<<16 * (LMAX + 1) * (LMAX + 1), 256, 0, stream>>>(FF, W, inv, out);
}